// Autoencoder_double_33389075759555
// MI455X (gfx1250) — compile-verified
//
#include <hip/hip_runtime.h>
#include <hip/hip_bf16.h>

// CDNA5 / gfx1250 autoencoder + forward-mode Jacobians using fp32 WMMA.
// Per 16-sample tile (one wave32):
//   pre1 = WMMA(q_tile[16x4 pad], W1^T[4x16]) + b1          (1 wmma)
//   h1,s1 = softplus/sigmoid elementwise (C/D layout, raw v_exp/v_log/v_rcp)
//   t1_j = s1 * W1[:,j]                                     (tangent seed)
//   LDS-transpose {h1,t1_0,t1_1} -> A layout, 3x chained-K16 WMMA vs W2^T
//   h2,s2; t2_j = s2 * u2_j
//   LDS-transpose {h2,t2_0,t2_1} -> 3x chained-K16 WMMA vs W3^T (N padded)
// Encoder then decoder (input = theta tile re-read through LDS).
// Analytic Jacobian computed on lanes 0..15 which hold q.

typedef __attribute__((ext_vector_type(2))) float v2f;
typedef __attribute__((ext_vector_type(8))) float v8f;

#define TSTR 18  // padded row stride (floats) of the 16x16 LDS transpose tile

__device__ __forceinline__ v8f wmma_k4(v2f a, v2f b, v8f c) {
  return __builtin_amdgcn_wmma_f32_16x16x4_f32(false, a, false, b, (short)0, c,
                                               false, false);
}

__device__ __forceinline__ v8f wmma_k16(const v2f* a, const v2f* b, v8f c) {
#pragma unroll
  for (int i = 0; i < 4; ++i)
    c = __builtin_amdgcn_wmma_f32_16x16x4_f32(false, a[i], false, b[i],
                                              (short)0, c, false, false);
  return c;
}

// Store a 16x16 tile held in C/D layout (M = vgpr + 8*(lane/16), N = lane%16).
__device__ __forceinline__ void tile_store(float* buf, int lane, v8f t) {
  const int n = lane & 15;
  const int mh = (lane >> 4) << 3;
#pragma unroll
  for (int r = 0; r < 8; ++r) buf[(mh + r) * TSTR + n] = t[r];
}

// Load the tile back in A layout: lane m holds K = 4c + 2*(lane/16) + {0,1}.
__device__ __forceinline__ void tile_loadA(const float* buf, int lane, v2f* a) {
  const int m = lane & 15;
  const int kh = (lane >> 4) << 1;
#pragma unroll
  for (int c = 0; c < 4; ++c) {
    const int kb = 4 * c + kh;
    a[c][0] = buf[m * TSTR + kb + 0];
    a[c][1] = buf[m * TSTR + kb + 1];
  }
}

// Stable softplus + sigmoid on an 8-reg tile, using raw hardware trans ops.
// Argument of v_log_f32 is 1+exp(-|x|) in (1,2]: no denorm/inf fixups needed,
// so bypass the guarded __logf/__expf expansions entirely.
__device__ __forceinline__ void act8(v8f x, v8f& h, v8f& s) {
  const float LOG2E = 1.4426950408889634f;
  const float LN2 = 0.6931471805599453f;
#pragma unroll
  for (int r = 0; r < 8; ++r) {
    float v = x[r];
    float e = __builtin_amdgcn_exp2f(-__builtin_fabsf(v) * LOG2E);  // exp(-|v|)
    float p = 1.0f + e;
    float d = __builtin_amdgcn_rcpf(p);           // sigmoid(|v|)
    h[r] = __builtin_fmaxf(v, 0.0f) + LN2 * __builtin_amdgcn_logf(p);
    s[r] = (v >= 0.0f) ? d : e * d;               // sigmoid(v)
  }
}

struct Net {
  v2f B1;            // W1^T as 4x16 B-matrix (K=2 real, rest zero)
  v2f B2[4];         // W2^T as 16x16 B-matrix, 4 K-chunks
  v2f B3[4];         // W3^T as 16x16 B-matrix (N=2 real, rest zero)
  v8f C1, C2, C3;    // bias tiles (broadcast over M)
  float w1c0, w1c1;  // W1 columns for tangent seeding (per-lane n)
};

__device__ __forceinline__ void net_load(Net& R, int lane, const float* w1,
                                         const float* b1, const float* w2,
                                         const float* b2, const float* w3,
                                         const float* b3) {
  const int n = lane & 15;
  const bool lo = lane < 16;
  const int kh = (lane >> 4) << 1;
  R.B1[0] = lo ? w1[n * 2 + 0] : 0.0f;
  R.B1[1] = lo ? w1[n * 2 + 1] : 0.0f;
#pragma unroll
  for (int c = 0; c < 4; ++c) {
    const int kb = 4 * c + kh;
    R.B2[c][0] = w2[n * 16 + kb + 0];
    R.B2[c][1] = w2[n * 16 + kb + 1];
    R.B3[c][0] = (n < 2) ? w3[n * 16 + kb + 0] : 0.0f;
    R.B3[c][1] = (n < 2) ? w3[n * 16 + kb + 1] : 0.0f;
  }
  const float c1 = b1[n];
  const float c2 = b2[n];
  const float c3 = (n < 2) ? b3[n] : 0.0f;
#pragma unroll
  for (int r = 0; r < 8; ++r) {
    R.C1[r] = c1;
    R.C2[r] = c2;
    R.C3[r] = c3;
  }
  R.w1c0 = w1[n * 2 + 0];
  R.w1c1 = w1[n * 2 + 1];
}

// One MLP forward + 2-column forward-mode Jacobian for a 16-sample tile.
// Ain: layer-1 A operand (16x4, K padded). Outputs in C/D layout (N=0,1 valid).
__device__ __forceinline__ void mlp_jac(const Net& R, v2f Ain, float* buf,
                                        int lane, v8f& out, v8f& j0, v8f& j1) {
  const v8f z = {0.f, 0.f, 0.f, 0.f, 0.f, 0.f, 0.f, 0.f};
  v8f pre1 = wmma_k4(Ain, R.B1, R.C1);
  v8f h1, s1;
  act8(pre1, h1, s1);
  v8f t10 = s1 * R.w1c0;  // d pre-act / d in_0 pushed through softplus'
  v8f t11 = s1 * R.w1c1;

  v2f a[4];
  tile_store(buf, lane, h1);
  tile_loadA(buf, lane, a);
  v8f pre2 = wmma_k16(a, R.B2, R.C2);
  tile_store(buf, lane, t10);
  tile_loadA(buf, lane, a);
  v8f u20 = wmma_k16(a, R.B2, z);
  tile_store(buf, lane, t11);
  tile_loadA(buf, lane, a);
  v8f u21 = wmma_k16(a, R.B2, z);

  v8f h2, s2;
  act8(pre2, h2, s2);
  v8f t20 = s2 * u20;
  v8f t21 = s2 * u21;

  tile_store(buf, lane, h2);
  tile_loadA(buf, lane, a);
  out = wmma_k16(a, R.B3, R.C3);
  tile_store(buf, lane, t20);
  tile_loadA(buf, lane, a);
  j0 = wmma_k16(a, R.B3, z);
  tile_store(buf, lane, t21);
  tile_loadA(buf, lane, a);
  j1 = wmma_k16(a, R.B3, z);
}

__global__ __launch_bounds__(256) void ae_jac_wmma_kernel(
    const float* __restrict__ q, const float* __restrict__ ew1,
    const float* __restrict__ eb1, const float* __restrict__ ew2,
    const float* __restrict__ eb2, const float* __restrict__ ew3,
    const float* __restrict__ eb3, const float* __restrict__ dw1,
    const float* __restrict__ db1, const float* __restrict__ dw2,
    const float* __restrict__ db2, const float* __restrict__ dw3,
    const float* __restrict__ db3, float* __restrict__ theta_o,
    float* __restrict__ jh_o, float* __restrict__ qhat_o,
    float* __restrict__ jd_o, float* __restrict__ jana_o, int N) {
  __shared__ __align__(16) float lds[8 * 16 * TSTR];  // per-wave transpose tile
  const int lane = threadIdx.x & 31;
  const int wave = threadIdx.x >> 5;
  float* buf = lds + wave * (16 * TSTR);

  Net E, D;
  net_load(E, lane, ew1, eb1, ew2, eb2, ew3, eb3);
  net_load(D, lane, dw1, db1, dw2, db2, dw3, db3);

  const int nTiles = (N + 15) >> 4;
  const int waveG = blockIdx.x * 8 + wave;
  const int nWaves = gridDim.x * 8;
  const int m = lane & 15;
  const int mh = (lane >> 4) << 3;

  for (int t = waveG; t < nTiles; t += nWaves) {
    const int base = t << 4;

    // ---- load q tile into A layout (K=2 real, K=2..3 zero) ----
    const int s = base + m;
    const int sc = (s < N) ? s : (N - 1);
    v2f Aq = {0.f, 0.f};
    float q0 = 0.f, q1 = 0.f;
    if (lane < 16) {
      q0 = q[2 * sc + 0];
      q1 = q[2 * sc + 1];
      Aq[0] = q0;
      Aq[1] = q1;
    }

    // ---- encoder ----
    v8f th, je0, je1;
    mlp_jac(E, Aq, buf, lane, th, je0, je1);

    // theta (C/D layout) -> A layout for the decoder (only K=0,1 valid)
    tile_store(buf, lane, th);
    v2f At = {0.f, 0.f};
    if (lane < 16) {
      At[0] = buf[m * TSTR + 0];
      At[1] = buf[m * TSTR + 1];
    }

    // ---- decoder ----
    v8f qh, jd0, jd1;
    mlp_jac(D, At, buf, lane, qh, jd0, jd1);

    // ---- stores: valid output columns live on lanes with n < 2 ----
    const int n = lane & 15;
    if (n < 2) {
#pragma unroll
      for (int r = 0; r < 8; ++r) {
        const int ss = base + mh + r;
        if (ss < N) {
          theta_o[ss * 2 + n] = th[r];
          qhat_o[ss * 2 + n] = qh[r];
          jh_o[ss * 4 + n * 2 + 0] = je0[r];
          jh_o[ss * 4 + n * 2 + 1] = je1[r];
          jd_o[ss * 4 + n * 2 + 0] = jd0[r];
          jd_o[ss * 4 + n * 2 + 1] = jd1[r];
        }
      }
    }

    // ---- analytic Jacobian (lanes 0..15 hold this tile's q) ----
    if (lane < 16 && s < N) {
      const float r2 = q0 * q0 + q1 * q1;
      const float inv = __builtin_amdgcn_rcpf(r2);          // atan2 grad (no eps)
      const float invs = __builtin_amdgcn_rsqf(r2 + 1e-8f); // d sqrt(r2+eps)
      float4 ja;
      ja.x = -q1 * inv;  // d theta0 / d q0
      ja.y = q0 * inv;   // d theta0 / d q1
      ja.z = q0 * invs;  // d theta1 / d q0
      ja.w = q1 * invs;  // d theta1 / d q1
      *(float4*)(jana_o + (size_t)s * 4) = ja;
    }
  }
}

extern "C" void kernel_launch(void* const* d_in, const int* in_sizes, int n_in,
                              void* d_out, int out_size, void* d_ws,
                              size_t ws_size, hipStream_t stream) {
  (void)n_in;
  (void)out_size;
  (void)d_ws;
  (void)ws_size;
  const float* q = (const float*)d_in[0];
  const float* ew1 = (const float*)d_in[1];
  const float* eb1 = (const float*)d_in[2];
  const float* ew2 = (const float*)d_in[3];
  const float* eb2 = (const float*)d_in[4];
  const float* ew3 = (const float*)d_in[5];
  const float* eb3 = (const float*)d_in[6];
  const float* dw1 = (const float*)d_in[7];
  const float* db1 = (const float*)d_in[8];
  const float* dw2 = (const float*)d_in[9];
  const float* db2 = (const float*)d_in[10];
  const float* dw3 = (const float*)d_in[11];
  const float* db3 = (const float*)d_in[12];

  const int N = in_sizes[0] / 2;
  float* out = (float*)d_out;
  float* theta_o = out;                  // [N,2]
  float* jh_o = out + (size_t)2 * N;     // [N,2,2]
  float* qhat_o = out + (size_t)6 * N;   // [N,2]
  float* jd_o = out + (size_t)8 * N;     // [N,2,2]
  float* jana_o = out + (size_t)12 * N;  // [N,2,2]

  const int nTiles = (N + 15) / 16;
  int blocks = 512;                        // persistent-ish; ~15 tiles/wave
  const int maxBlocks = (nTiles + 7) / 8;  // >= 1 tile per wave
  if (blocks > maxBlocks) blocks = maxBlocks;
  if (blocks < 1) blocks = 1;

  ae_jac_wmma_kernel<<<blocks, 256, 0, stream>>>(
      q, ew1, eb1, ew2, eb2, ew3, eb3, dw1, db1, dw2, db2, dw3, db3, theta_o,
      jh_o, qhat_o, jd_o, jana_o, N);
}